// SasakiProjectionMemory_75419625718530
// MI455X (gfx1250) — compile-verified
//
#include <hip/hip_runtime.h>

// Sasaki projection memory scan for MI455X (gfx1250, wave32).
// One workgroup (256 threads = 8 waves) per (B,H) chain; U,V resident in LDS
// (~274KB of the 320KB WGP LDS); matvecs via v_wmma_f32_16x16x4_f32.
// Per step: 1 projection GEMM + 2 apply GEMMs (the second projection is
// replaced by an exact rank-1 coefficient patch computed in the reduction).

#define EPSV     1e-6f
#define Dd       256
#define Rr       64
#define Tt       512
#define STRIDE   130         // 2*Rr + 2 pad floats -> conflict-free both orientations
#define NTHREADS 256

typedef __attribute__((ext_vector_type(2))) float v2f;
typedef __attribute__((ext_vector_type(8))) float v8f;

__device__ __forceinline__ v8f wmma4(v2f a, v2f b, v8f c) {
  // D(16x16,f32) = A(16x4,f32) x B(4x16,f32) + C
  return __builtin_amdgcn_wmma_f32_16x16x4_f32(false, a, false, b, (short)0, c,
                                               false, false);
}

// proj: D = stacked[Ur^T;Ui^T](128 x 256) * [xr xi 0...](256 x 16)
// kbuf = [kr(256) | ki(256) | zeros(256)]; lane picks its column base once.
__device__ __forceinline__ void gemm_proj(const float* __restrict__ Umat,
                                          const float* __restrict__ kbuf,
                                          float* __restrict__ pA,   // col 0
                                          float* __restrict__ pB,   // col 1
                                          int wave, int lane) {
  const int nCol  = lane & 15;
  const int kHalf = (lane >> 4) * 2;
  const int mBase = wave * 16;
  const int mRow  = mBase + (lane & 15);
  const int basep = (nCol < 2) ? nCol * Dd : 2 * Dd;   // selects kr / ki / zeros
  v8f acc = {};
#pragma unroll 8
  for (int kb = 0; kb < Dd; kb += 4) {
    const int ka = kb + kHalf;
    v2f a;
    a.x = Umat[ka * STRIDE + mRow];
    a.y = Umat[(ka + 1) * STRIDE + mRow];
    v2f b;
    b.x = kbuf[basep + ka];
    b.y = kbuf[basep + ka + 1];
    acc = wmma4(a, b, acc);
  }
  if (nCol < 2) {
    float* dst = nCol ? pB : pA;
    const int mOff = mBase + ((lane >> 4) ? 8 : 0);
#pragma unroll
    for (int j = 0; j < 8; ++j) dst[mOff + j] = acc[j];
  }
}

// apply: D = [Ar|Ai](256 x 128) * bapp(128 x 16); wave w owns M-tiles 2w, 2w+1.
// bapp = [col0(128) | col1(128) | zeros(128)], col0=[cr;-ci], col1=[ci;cr].
__device__ __forceinline__ void gemm_apply(const float* __restrict__ Amat,
                                           const float* __restrict__ bapp,
                                           float* __restrict__ outR,
                                           float* __restrict__ outI,
                                           int wave, int lane) {
  const int nCol  = lane & 15;
  const int kHalf = (lane >> 4) * 2;
  const int basea = (nCol < 2) ? nCol * (2 * Rr) : 2 * (2 * Rr);
#pragma unroll
  for (int tile = 0; tile < 2; ++tile) {
    const int mBase = (wave * 2 + tile) * 16;
    const int mRow  = mBase + (lane & 15);
    v8f acc = {};
#pragma unroll 8
    for (int kb = 0; kb < 2 * Rr; kb += 4) {
      const int ka = kb + kHalf;
      v2f a;
      a.x = Amat[mRow * STRIDE + ka];
      a.y = Amat[mRow * STRIDE + ka + 1];
      v2f b;
      b.x = bapp[basea + ka];
      b.y = bapp[basea + ka + 1];
      acc = wmma4(a, b, acc);
    }
    if (nCol < 2) {
      float* dst = nCol ? outI : outR;
      const int mOff = mBase + ((lane >> 4) ? 8 : 0);
#pragma unroll
      for (int j = 0; j < 8; ++j) dst[mOff + j] = acc[j];
    }
  }
}

__global__ void __launch_bounds__(NTHREADS, 1)
sasaki_scan_kernel(const float* __restrict__ k_seq,
                   const float* __restrict__ v_seq,
                   const float* __restrict__ gamma_seq,
                   const float* __restrict__ U0,
                   const float* __restrict__ V0,
                   const int*   __restrict__ next_slot0,
                   float* __restrict__ out) {
  extern __shared__ float smem[];
  float* U    = smem;                  // 256*130
  float* V    = U + Dd * STRIDE;       // 256*130
  float* kbuf = V + Dd * STRIDE;       // 768: kr | ki | zeros
  float* vr   = kbuf + 3 * Dd;         // 256
  float* vi   = vr + Dd;               // 256
  float* pA   = vi + Dd;               // 128
  float* pB   = pA + 2 * Rr;           // 128
  float* bapp = pB + 2 * Rr;           // 384: col0 | col1 | zeros
  float* prr  = bapp + 3 * 2 * Rr;     // 256 projection (U c) real
  float* pri  = prr + Dd;              // 256 imag
  float* kpr  = pri + Dd;              // 256 residual real
  float* kpi  = kpr + Dd;              // 256 residual imag
  float* yrL  = kpi + Dd;              // 256 output real
  float* yiL  = yrL + Dd;              // 256 output imag
  float* red  = yiL + Dd;              // 32 (8 waves x 3 sums)
  float* bc   = red + 32;              // 8 broadcast slots

  const int tid  = threadIdx.x;
  const int lane = tid & 31;
  const int wave = tid >> 5;
  const int bh   = blockIdx.x;

  // ---- init: U0/V0 -> LDS (deinterleaved, [d][Ur(64)|Ui(64)]), zero panels ----
  const float* U0p = U0 + (size_t)bh * Dd * Rr * 2;
  const float* V0p = V0 + (size_t)bh * Dd * Rr * 2;
  for (int i = tid; i < Dd * Rr; i += NTHREADS) {
    const int dd = i >> 6, rr = i & 63;
    U[dd * STRIDE + rr]      = U0p[2 * i];
    U[dd * STRIDE + Rr + rr] = U0p[2 * i + 1];
    V[dd * STRIDE + rr]      = V0p[2 * i];
    V[dd * STRIDE + Rr + rr] = V0p[2 * i + 1];
  }
  kbuf[2 * Dd + tid] = 0.0f;           // zero column panel for proj
  if (tid < 2 * Rr) bapp[2 * (2 * Rr) + tid] = 0.0f;  // zero panel for apply
  int slot = next_slot0[bh] & (Rr - 1);

  const float2* ksp = (const float2*)(k_seq + (size_t)bh * Tt * Dd * 2);
  const float2* vsp = (const float2*)(v_seq + (size_t)bh * Tt * Dd * 2);
  const float*  gsp = gamma_seq + (size_t)bh * Tt;
  float2*       osp = (float2*)(out + (size_t)bh * Tt * Dd * 2);

  // prime the prefetch registers (t = 0)
  float2 pk = ksp[tid];
  float2 pv = vsp[tid];
  float  pg = gsp[0];
  __syncthreads();

#pragma unroll 1
  for (int t = 0; t < Tt; ++t) {
    // ---- stage current step; prefetch t+1 into regs, t+4 into L2 ----
    kbuf[tid]      = pk.x;
    kbuf[Dd + tid] = pk.y;
    vr[tid] = pv.x; vi[tid] = pv.y;
    if (tid == 0) bc[4] = pg;
    const int tn = (t + 1 < Tt) ? t + 1 : t;
    pk = ksp[(size_t)tn * Dd + tid];
    pv = vsp[(size_t)tn * Dd + tid];
    pg = gsp[tn];
    const int tf = (t + 4 < Tt) ? t + 4 : t;
    __builtin_prefetch(&ksp[(size_t)tf * Dd + tid], 0, 0);  // global_prefetch_b8
    __builtin_prefetch(&vsp[(size_t)tf * Dd + tid], 0, 0);
    __syncthreads();

    // ---- coef = U^H k ----
    gemm_proj(U, kbuf, pA, pB, wave, lane);
    __syncthreads();
    if (tid < Rr) {
      const float cr = pA[tid] + pB[Rr + tid];   // Ur^T xr + Ui^T xi
      const float ci = pB[tid] - pA[Rr + tid];   // Ur^T xi - Ui^T xr
      bapp[tid]            = cr;                 // col0 = [cr; -ci]
      bapp[Rr + tid]       = -ci;
      bapp[2 * Rr + tid]   = ci;                 // col1 = [ci; cr]
      bapp[3 * Rr + tid]   = cr;
    }
    __syncthreads();

    // ---- p = U c ; k_perp = k - p ; fused reduction of 3 dot products ----
    gemm_apply(U, bapp, prr, pri, wave, lane);
    __syncthreads();
    const float krv  = kbuf[tid], kiv = kbuf[Dd + tid];
    const float kprv = krv - prr[tid];
    const float kpiv = kiv - pri[tid];
    kpr[tid] = kprv; kpi[tid] = kpiv;
    float s0 = kprv * kprv + kpiv * kpiv;   // ||k_perp||^2
    float s1 = kprv * krv + kpiv * kiv;     // Re(k_perp^H k)
    float s2 = kprv * kiv - kpiv * krv;     // Im(k_perp^H k)
#pragma unroll
    for (int m = 16; m > 0; m >>= 1) {
      s0 += __shfl_xor(s0, m);
      s1 += __shfl_xor(s1, m);
      s2 += __shfl_xor(s2, m);
    }
    if (lane == 0) { red[wave] = s0; red[8 + wave] = s1; red[16 + wave] = s2; }
    __syncthreads();

    if (tid == 0) {
      float sq = 0.f, d1 = 0.f, d2 = 0.f;
#pragma unroll
      for (int w = 0; w < 8; ++w) { sq += red[w]; d1 += red[8 + w]; d2 += red[16 + w]; }
      const float nrm = sqrtf(fmaxf(sq, 1e-24f));
      const float inv = 1.0f / sqrtf(fmaxf(sq, EPSV * EPSV));
      const float wr  = (nrm > EPSV) ? 1.0f : 0.0f;
      bc[0] = inv; bc[1] = wr; bc[2] = fminf(fmaxf(bc[4], 0.0f), 1.0f);
      if (wr != 0.0f) {
        // exact coefficient patch: c_new[slot] = u_new^H k (others unchanged)
        const float crs = inv * d1, cis = inv * d2;
        bapp[slot]          = crs;
        bapp[Rr + slot]     = -cis;
        bapp[2 * Rr + slot] = cis;
        bapp[3 * Rr + slot] = crs;
      }
    }
    __syncthreads();
    const float inv = bc[0];
    const bool  wrt = bc[1] != 0.0f;
    const float gc  = bc[2];

    // ---- update: V *= gamma everywhere; write column `slot` of U and V ----
    {
      float* Vrow = V + tid * STRIDE;
#pragma unroll 8
      for (int c = 0; c < 2 * Rr; ++c) Vrow[c] *= gc;
      if (wrt) {
        float* Urow = U + tid * STRIDE;
        Urow[slot]      = kpr[tid] * inv;
        Urow[Rr + slot] = kpi[tid] * inv;
        Vrow[slot]      = vr[tid] * gc;   // column written then scaled by gamma
        Vrow[Rr + slot] = vi[tid] * gc;
      }
    }
    __syncthreads();

    // ---- y = V_new * c_new ----
    gemm_apply(V, bapp, yrL, yiL, wave, lane);
    __syncthreads();

    float2 yo; yo.x = yrL[tid]; yo.y = yiL[tid];
    osp[(size_t)t * Dd + tid] = yo;        // coalesced 2KB store

    slot = (slot + 1) & (Rr - 1);
  }
}

static constexpr size_t SMEM_BYTES =
    (size_t)(2 * Dd * STRIDE + 3 * Dd + 2 * Dd + 2 * (2 * Rr) + 3 * (2 * Rr) +
             6 * Dd + 32 + 8) * sizeof(float);   // ~280 KB < 320 KB WGP LDS

extern "C" void kernel_launch(void* const* d_in, const int* in_sizes, int n_in,
                              void* d_out, int out_size, void* d_ws, size_t ws_size,
                              hipStream_t stream) {
  const float* k_seq      = (const float*)d_in[0];
  const float* v_seq      = (const float*)d_in[1];
  const float* gamma_seq  = (const float*)d_in[2];
  const float* U0         = (const float*)d_in[3];
  const float* V0         = (const float*)d_in[4];
  const int*   next_slot0 = (const int*)d_in[5];
  // d_in[6] (filled0) does not affect the output.
  float* out = (float*)d_out;

  hipFuncSetAttribute((const void*)sasaki_scan_kernel,
                      hipFuncAttributeMaxDynamicSharedMemorySize,
                      (int)SMEM_BYTES);

  sasaki_scan_kernel<<<64, NTHREADS, SMEM_BYTES, stream>>>(
      k_seq, v_seq, gamma_seq, U0, V0, next_slot0, out);
}